// GCN_37675453120656
// MI455X (gfx1250) — compile-verified
//
#include <hip/hip_runtime.h>

typedef __attribute__((ext_vector_type(2))) float v2f;
typedef __attribute__((ext_vector_type(8))) float v8f;

// ---------------------------------------------------------------- utilities
__global__ void fill_f32(float* __restrict__ p, float v, int n) {
    int i = blockIdx.x * blockDim.x + threadIdx.x;
    if (i < n) p[i] = v;
}

// deg[i] starts at 1.0 (self loop); add 1 per in-edge
__global__ void deg_edges(const int* __restrict__ dst, float* __restrict__ deg, int nE) {
    int e = blockIdx.x * blockDim.x + threadIdx.x;
    if (e < nE) atomicAdd(&deg[dst[e]], 1.0f);
}

// in-place deg -> deg^{-1/2}   (deg >= 1 always, so no clamp needed)
__global__ void deg_to_dis(float* __restrict__ d, int n) {
    int i = blockIdx.x * blockDim.x + threadIdx.x;
    if (i < n) d[i] = rsqrtf(d[i]);
}

// ---------------------------------------------------------------- WMMA GEMM
// Y[N,F] = X[N,K] @ W[K,F] (+ bias), F = FT*16, K % 4 == 0.
// One wave computes a 16-row x F tile with V_WMMA_F32_16X16X4_F32.
// A frag (16x4 f32): lane<16 -> row=lane, K=k0..k0+1 ; lane>=16 -> row=lane-16, K=k0+2..k0+3
// B frag (4x16 f32): same K split across the two half-waves, N = lane&15
// C/D  (16x16 f32): vgpr v, lane<16 -> row v ; lane>=16 -> row v+8 ; col = lane&15
template<int FT>
__global__ void wmma_gemm(const float* __restrict__ X, const float* __restrict__ W,
                          const float* __restrict__ Bias, float* __restrict__ Y,
                          int N, int K, int addBias) {
    const int F    = FT * 16;
    int wave = (blockIdx.x * blockDim.x + threadIdx.x) >> 5;
    int lane = threadIdx.x & 31;
    int row0 = wave << 4;
    if (row0 >= N) return;                 // uniform per wave -> EXEC stays all-ones below
    int half = lane >> 4;                  // which K-pair this half-wave holds
    int l16  = lane & 15;
    int arow = row0 + l16; if (arow > N - 1) arow = N - 1;   // clamp: row-independent math

    v8f acc[FT];
    v8f z = {0.f, 0.f, 0.f, 0.f, 0.f, 0.f, 0.f, 0.f};
#pragma unroll
    for (int t = 0; t < FT; ++t) acc[t] = z;

    for (int k0 = 0; k0 < K; k0 += 4) {
        int ka = k0 + half * 2;
        v2f a;
        a.x = X[arow * K + ka];
        a.y = X[arow * K + ka + 1];
#pragma unroll
        for (int t = 0; t < FT; ++t) {
            int col = t * 16 + l16;
            v2f b;
            b.x = W[ka * F + col];
            b.y = W[(ka + 1) * F + col];
            acc[t] = __builtin_amdgcn_wmma_f32_16x16x4_f32(
                /*neg_a=*/false, a, /*neg_b=*/false, b,
                /*c_mod=*/(short)0, acc[t], /*reuse_a=*/false, /*reuse_b=*/false);
        }
    }

#pragma unroll
    for (int t = 0; t < FT; ++t) {
        int col = t * 16 + l16;
        float bb = addBias ? Bias[col] : 0.f;
#pragma unroll
        for (int v = 0; v < 8; ++v) {
            int r = row0 + half * 8 + v;
            if (r < N) Y[r * F + col] = acc[t][v] + bb;
        }
    }
}

// ---------------------------------------------------------------- aggregation
// self-loop contribution: AGG[i,f] = H[i,f] * dis[i]^2
__global__ void agg_init(const float* __restrict__ H, const float* __restrict__ dis,
                         float* __restrict__ AGG, int N, int F) {
    int idx = blockIdx.x * blockDim.x + threadIdx.x;
    if (idx < N * F) {
        float d = dis[idx / F];
        AGG[idx] = H[idx] * d * d;
    }
}

// one wave per edge, lane strides features; atomics land in L2 (arrays << 192MB)
__global__ void edge_scatter(const int* __restrict__ src, const int* __restrict__ dst,
                             const float* __restrict__ dis, const float* __restrict__ H,
                             float* __restrict__ AGG, int nE, int F) {
    int e    = (blockIdx.x * blockDim.x + threadIdx.x) >> 5;
    int lane = threadIdx.x & 31;
    if (e >= nE) return;
    int   s = src[e];
    int   d = dst[e];
    float w = dis[s] * dis[d];
    for (int f = lane; f < F; f += 32)
        atomicAdd(&AGG[d * F + f], H[s * F + f] * w);
}

__global__ void bias_relu(float* __restrict__ A, const float* __restrict__ b, int N, int F) {
    int idx = blockIdx.x * blockDim.x + threadIdx.x;
    if (idx < N * F) {
        float v = A[idx] + b[idx % F];
        A[idx] = v > 0.f ? v : 0.f;
    }
}

// ---------------------------------------------------------------- pooling
// values are post-ReLU (>= 0), so uint atomicMax is order-correct and init=0 matches
__global__ void pool_nodes(const float* __restrict__ H, const int* __restrict__ batch,
                           float* __restrict__ gmax, float* __restrict__ gsum,
                           float* __restrict__ cnt, int N, int F) {
    int i    = (blockIdx.x * blockDim.x + threadIdx.x) >> 5;
    int lane = threadIdx.x & 31;
    if (i >= N) return;
    int g = batch[i];
    for (int f = lane; f < F; f += 32) {
        float v = H[i * F + f];
        atomicMax((unsigned int*)&gmax[g * F + f], __float_as_uint(v));
        atomicAdd(&gsum[g * F + f], v);
    }
    if (lane == 0) atomicAdd(&cnt[g], 1.0f);
}

__global__ void build_pooled(const float* __restrict__ gmax, const float* __restrict__ gsum,
                             const float* __restrict__ cnt, float* __restrict__ pooled,
                             int G, int F) {
    int idx = blockIdx.x * blockDim.x + threadIdx.x;
    if (idx >= G * 2 * F) return;
    int g = idx / (2 * F);
    int f = idx % (2 * F);
    float v;
    if (f < F) {
        v = gmax[g * F + f];
    } else {
        float c = cnt[g];
        v = gsum[g * F + (f - F)] / (c > 1.f ? c : 1.f);
    }
    pooled[idx] = v;
}

// ---------------------------------------------------------------- host entry
extern "C" void kernel_launch(void* const* d_in, const int* in_sizes, int n_in,
                              void* d_out, int out_size, void* d_ws, size_t ws_size,
                              hipStream_t stream) {
    const float* x     = (const float*)d_in[0];
    const int*   ei    = (const int*)d_in[1];
    const int*   batch = (const int*)d_in[2];
    const float* W1    = (const float*)d_in[3];
    const float* b1    = (const float*)d_in[4];
    const float* W2    = (const float*)d_in[5];
    const float* b2    = (const float*)d_in[6];
    const float* W3    = (const float*)d_in[7];
    const float* b3    = (const float*)d_in[8];
    const float* Wout  = (const float*)d_in[9];
    const float* bout  = (const float*)d_in[10];
    float*       out   = (float*)d_out;

    const int N  = in_sizes[2];           // 100000 nodes
    const int E  = in_sizes[1] / 2;       // 3.2M edges
    const int K0 = in_sizes[0] / N;       // 128 input features
    const int F1 = in_sizes[4];           // 32
    const int F2 = in_sizes[6];           // 48
    const int F3 = in_sizes[8];           // 64
    const int Fo = in_sizes[10];          // 32
    const int G  = out_size / Fo;         // 512 graphs

    const int* src = ei;
    const int* dst = ei + E;

    // ---- workspace carving (256B aligned)
    char*  ws  = (char*)d_ws;
    size_t off = 0;
    auto carve = [&](size_t bytes) -> float* {
        float* p = (float*)(ws + off);
        off += (bytes + 255) & ~(size_t)255;
        return p;
    };
    float* dis    = carve((size_t)N * 4);            // deg -> dis in place
    float* bufT   = carve((size_t)N * F3 * 4);       // GEMM outputs (max width 64)
    float* bufA   = carve((size_t)N * F3 * 4);       // agg / layer activations
    float* bufB   = carve((size_t)N * F3 * 4);
    float* gmax   = carve((size_t)G * F3 * 4);
    float* gsum   = carve((size_t)G * F3 * 4);
    float* cnt    = carve((size_t)G * 4);
    float* pooled = carve((size_t)G * 2 * F3 * 4);

    const int BLK = 256;
    auto grid1   = [&](long long n) { return dim3((unsigned)((n + BLK - 1) / BLK)); };
    auto gridWv  = [&](long long waves) { return dim3((unsigned)((waves * 32 + BLK - 1) / BLK)); };
    auto gemmGrid = [&](int rows) { return gridWv((rows + 15) / 16); };

    // ---- degrees & symmetric norm
    fill_f32<<<grid1(N), BLK, 0, stream>>>(dis, 1.0f, N);          // self loop
    deg_edges<<<grid1(E), BLK, 0, stream>>>(dst, dis, E);
    deg_to_dis<<<grid1(N), BLK, 0, stream>>>(dis, N);

    // ---- layer 1: 128 -> 32
    wmma_gemm<2><<<gemmGrid(N), BLK, 0, stream>>>(x, W1, b1, bufT, N, K0, 0);
    agg_init<<<grid1((long long)N * F1), BLK, 0, stream>>>(bufT, dis, bufA, N, F1);
    edge_scatter<<<gridWv(E), BLK, 0, stream>>>(src, dst, dis, bufT, bufA, E, F1);
    bias_relu<<<grid1((long long)N * F1), BLK, 0, stream>>>(bufA, b1, N, F1);

    // ---- layer 2: 32 -> 48
    wmma_gemm<3><<<gemmGrid(N), BLK, 0, stream>>>(bufA, W2, b2, bufT, N, F1, 0);
    agg_init<<<grid1((long long)N * F2), BLK, 0, stream>>>(bufT, dis, bufB, N, F2);
    edge_scatter<<<gridWv(E), BLK, 0, stream>>>(src, dst, dis, bufT, bufB, E, F2);
    bias_relu<<<grid1((long long)N * F2), BLK, 0, stream>>>(bufB, b2, N, F2);

    // ---- layer 3: 48 -> 64
    wmma_gemm<4><<<gemmGrid(N), BLK, 0, stream>>>(bufB, W3, b3, bufT, N, F2, 0);
    agg_init<<<grid1((long long)N * F3), BLK, 0, stream>>>(bufT, dis, bufA, N, F3);
    edge_scatter<<<gridWv(E), BLK, 0, stream>>>(src, dst, dis, bufT, bufA, E, F3);
    bias_relu<<<grid1((long long)N * F3), BLK, 0, stream>>>(bufA, b3, N, F3);

    // ---- pooling (max + mean per graph)
    fill_f32<<<grid1((long long)G * F3), BLK, 0, stream>>>(gmax, 0.0f, G * F3);
    fill_f32<<<grid1((long long)G * F3), BLK, 0, stream>>>(gsum, 0.0f, G * F3);
    fill_f32<<<grid1(G), BLK, 0, stream>>>(cnt, 0.0f, G);
    pool_nodes<<<gridWv(N), BLK, 0, stream>>>(bufA, batch, gmax, gsum, cnt, N, F3);
    build_pooled<<<grid1((long long)G * 2 * F3), BLK, 0, stream>>>(gmax, gsum, cnt, pooled, G, F3);

    // ---- final projection: [G,128] @ [128,32] + bout
    wmma_gemm<2><<<gemmGrid(G), BLK, 0, stream>>>(pooled, Wout, bout, out, G, 2 * F3, 1);
}